// Group_34144990003370
// MI455X (gfx1250) — compile-verified
//
#include <hip/hip_runtime.h>
#include <hip/hip_bf16.h>

// Problem constants (match setup_inputs: xyz [16,16384,3] f32, num_group=512, group_size=32)
#define NPTS    16384
#define NGROUP  512
#define GSIZE   32

typedef __attribute__((ext_vector_type(2))) float v2f;
typedef __attribute__((ext_vector_type(8))) float v8f;

// ---------------------------------------------------------------------------
// Kernel 1: Farthest Point Sampling. One block per batch, 1024 threads,
// 16 points per thread held in registers. Writes center coords (= xyz gathered
// at the FPS indices) into the `centers` region of d_out.
// ---------------------------------------------------------------------------
#define FPS_T   1024
#define FPS_PPT (NPTS / FPS_T)   // 16

__global__ __launch_bounds__(FPS_T)
void fps_kernel(const float* __restrict__ xyz, float* __restrict__ centers) {
    const int b = blockIdx.x;
    const int t = threadIdx.x;
    const float* pts = xyz + (size_t)b * NPTS * 3;

    float px[FPS_PPT], py[FPS_PPT], pz[FPS_PPT], dst[FPS_PPT];
#pragma unroll
    for (int k = 0; k < FPS_PPT; ++k) {
        int i = k * FPS_T + t;
        px[k] = pts[i * 3 + 0];
        py[k] = pts[i * 3 + 1];
        pz[k] = pts[i * 3 + 2];
        dst[k] = 1.0e10f;   // BIG
    }

    __shared__ float s_c[3];
    __shared__ int   s_far;
    __shared__ float s_wd[32];
    __shared__ int   s_wi[32];

    int farthest = 0;   // deterministic start at index 0

    for (int g = 0; g < NGROUP; ++g) {
        // Owner of current farthest point publishes its coords + emits center g.
        if (t == (farthest & (FPS_T - 1))) {
            int k = farthest >> 10;          // farthest / 1024
            s_c[0] = px[k]; s_c[1] = py[k]; s_c[2] = pz[k];
            float* co = centers + ((size_t)b * NGROUP + g) * 3;
            co[0] = px[k]; co[1] = py[k]; co[2] = pz[k];
        }
        __syncthreads();                     // barrier A
        const float cx = s_c[0], cy = s_c[1], cz = s_c[2];

        float bestd = -1.0f;
        int   besti = NPTS;
#pragma unroll
        for (int k = 0; k < FPS_PPT; ++k) {
            float dx = px[k] - cx, dy = py[k] - cy, dz = pz[k] - cz;
            float d  = dx * dx + dy * dy + dz * dz;     // ((dx^2+dy^2)+dz^2)
            float nd = fminf(dst[k], d);
            dst[k] = nd;
            int i = k * FPS_T + t;
            if (nd > bestd || (nd == bestd && i < besti)) { bestd = nd; besti = i; }
        }
        // wave32 argmax reduction (greater dist wins, ties -> lower index)
#pragma unroll
        for (int off = 16; off > 0; off >>= 1) {
            float od = __shfl_down(bestd, off, 32);
            int   oi = __shfl_down(besti, off, 32);
            if (od > bestd || (od == bestd && oi < besti)) { bestd = od; besti = oi; }
        }
        if ((t & 31) == 0) { s_wd[t >> 5] = bestd; s_wi[t >> 5] = besti; }
        __syncthreads();                     // barrier B
        if (t < 32) {
            bestd = s_wd[t]; besti = s_wi[t];
#pragma unroll
            for (int off = 16; off > 0; off >>= 1) {
                float od = __shfl_down(bestd, off, 32);
                int   oi = __shfl_down(besti, off, 32);
                if (od > bestd || (od == bestd && oi < besti)) { bestd = od; besti = oi; }
            }
            if (t == 0) s_far = besti;
        }
        __syncthreads();                     // barrier C
        farthest = s_far;
    }
}

// ---------------------------------------------------------------------------
// Kernel 2: kNN (top-32 by squared distance, exact top_k tie order) + grouping.
// One block per (batch, 16-center tile): 16*32 = 512 blocks, 256 threads.
// Distances for 16 centers x 16 points come from V_WMMA_F32_16X16X4_F32
// (K=4, z-padded with 0), staged through LDS in 512-point chunks.
// ---------------------------------------------------------------------------
#define CHUNK 512

__global__ __launch_bounds__(256)
void knn_group_kernel(const float* __restrict__ xyz,
                      const float* __restrict__ centers,
                      float* __restrict__ nbr_out) {
    const int b    = blockIdx.x >> 5;        // / (NGROUP/16 = 32)
    const int tile = blockIdx.x & 31;
    const int s0   = tile * 16;
    const int t    = threadIdx.x;
    const int lane = t & 31;
    const int wave = t >> 5;                 // 0..7

    __shared__ float sC[16][3];
    __shared__ float sQn[16];
    __shared__ float sDist[16][CHUNK];       // 32 KB
    __shared__ float sTopD[16][GSIZE];
    __shared__ int   sTopI[16][GSIZE];

    const float* pts = xyz + (size_t)b * NPTS * 3;
    const float* ctr = centers + ((size_t)b * NGROUP + s0) * 3;

    if (t < 16) {
        float x = ctr[t * 3 + 0], y = ctr[t * 3 + 1], z = ctr[t * 3 + 2];
        sC[t][0] = x; sC[t][1] = y; sC[t][2] = z;
        sQn[t] = x * x + y * y + z * z;      // |q|^2
#pragma unroll
        for (int j = 0; j < GSIZE; ++j) { sTopD[t][j] = 3.0e38f; sTopI[t][j] = 0x7fffffff; }
    }
    __syncthreads();

    // A operand (16x4 f32): lanes 0-15 -> (K0,K1)=(x,y) of center M=lane,
    // lanes 16-31 -> (K2,K3)=(z,0).
    const int mlo = lane & 15;
    v2f a;
    a.x = (lane < 16) ? sC[mlo][0] : sC[mlo][2];
    a.y = (lane < 16) ? sC[mlo][1] : 0.0f;

    for (int chunk = 0; chunk < NPTS / CHUNK; ++chunk) {
        const int base = chunk * CHUNK;

        // ---- Phase A: 16x16 distance tiles via WMMA -> LDS ----
#pragma unroll
        for (int it = 0; it < CHUNK / (8 * 16); ++it) {       // 4 tiles per wave
            const int col0 = wave * (CHUNK / 8) + it * 16;    // local column base
            const int p    = base + col0 + mlo;               // global point idx
            float x = pts[p * 3 + 0];
            float y = pts[p * 3 + 1];
            float z = pts[p * 3 + 2];
            float pn = x * x + y * y + z * z;                 // |p|^2

            // B operand (4x16 f32): lanes 0-15 -> rows (K0,K1)=(x,y) of point
            // N=lane, lanes 16-31 -> rows (K2,K3)=(z,0).
            v2f bb;
            bb.x = (lane < 16) ? x : z;
            bb.y = (lane < 16) ? y : 0.0f;

            v8f c = {};
            c = __builtin_amdgcn_wmma_f32_16x16x4_f32(
                    /*neg_a=*/false, a, /*neg_b=*/false, bb,
                    /*c_mod=*/(short)0, c, /*reuse_a=*/false, /*reuse_b=*/false);

            // D layout: VGPR j -> row M = j + 8*(lane>=16), col N = lane%16
            const int mrow0 = (lane < 16) ? 0 : 8;
            const int colL  = col0 + mlo;
#pragma unroll
            for (int j = 0; j < 8; ++j) {
                const int mm = mrow0 + j;
                sDist[mm][colL] = sQn[mm] + pn - 2.0f * c[j];
            }
        }
        __syncthreads();

        // ---- Phase B: per-center running sorted top-32 (dist, idx) ----
        if (t < 16) {
            float* myD = sTopD[t];
            int*   myI = sTopI[t];
            float worst  = myD[GSIZE - 1];
            int   worsti = myI[GSIZE - 1];
            for (int cI = 0; cI < CHUNK; ++cI) {
                float d  = sDist[t][cI];
                int   gi = base + cI;
                if (d < worst || (d == worst && gi < worsti)) {
                    int j = GSIZE - 1;
                    while (j > 0 && (myD[j - 1] > d ||
                                     (myD[j - 1] == d && myI[j - 1] > gi))) {
                        myD[j] = myD[j - 1]; myI[j] = myI[j - 1]; --j;
                    }
                    myD[j] = d; myI[j] = gi;
                    worst  = myD[GSIZE - 1];
                    worsti = myI[GSIZE - 1];
                }
            }
        }
        __syncthreads();
    }

    // ---- Phase C: gather neighborhoods, subtract center ----
    // 16 centers * 32 neighbors = 512 entries -> 2 per thread.
#pragma unroll
    for (int p = t; p < 16 * GSIZE; p += 256) {
        const int sl = p >> 5;               // local center
        const int j  = p & (GSIZE - 1);      // neighbor rank
        const int gi = sTopI[sl][j];
        const float ox = pts[gi * 3 + 0] - sC[sl][0];
        const float oy = pts[gi * 3 + 1] - sC[sl][1];
        const float oz = pts[gi * 3 + 2] - sC[sl][2];
        const size_t o = ((((size_t)b * NGROUP) + s0 + sl) * GSIZE + j) * 3;
        nbr_out[o + 0] = ox;
        nbr_out[o + 1] = oy;
        nbr_out[o + 2] = oz;
    }
}

// ---------------------------------------------------------------------------
extern "C" void kernel_launch(void* const* d_in, const int* in_sizes, int n_in,
                              void* d_out, int out_size, void* d_ws, size_t ws_size,
                              hipStream_t stream) {
    (void)n_in; (void)out_size; (void)d_ws; (void)ws_size;
    const float* xyz = (const float*)d_in[0];
    const int B = in_sizes[0] / (NPTS * 3);   // 16

    float* out     = (float*)d_out;
    // d_out layout: neighborhood [B,G,M,3] then center [B,G,3]
    float* centers = out + (size_t)B * NGROUP * GSIZE * 3;

    fps_kernel<<<B, FPS_T, 0, stream>>>(xyz, centers);
    knn_group_kernel<<<B * (NGROUP / 16), 256, 0, stream>>>(xyz, centers, out);
}